// LiquidStateMachine_73297911873840
// MI455X (gfx1250) — compile-verified
//
#include <hip/hip_runtime.h>

// ---------------------------------------------------------------------------
// Reference collapses to:  u = 0.01f * (x[524288,32] @ x2h[32,256]),  spk = 0.
// h2h (d_in[2]) is multiplied by a zero state -> dead input.
// Memory-bound problem (~1.09 GB traffic -> ~47us @ 23.3 TB/s); use the native
// fp32 matrix op V_WMMA_F32_16X16X4_F32 so precision matches the reference.
// ---------------------------------------------------------------------------

typedef __attribute__((ext_vector_type(2))) float v2f;
typedef __attribute__((ext_vector_type(8))) float v8f;

#define B_ROWS 524288
#define N_INP 32
#define N_HID 256
#define ROWS_PER_BLOCK 128   // 8 waves * 16 rows
#define LDS_PITCH 34         // 32 + 2 pad floats: kills LDS bank conflicts
#define SCALE 0.01f          // RC * DT

__global__ __launch_bounds__(256) void lsm_gemm_kernel(
    const float* __restrict__ x,      // [B, 32]
    const float* __restrict__ x2h,    // [32, 256]
    float* __restrict__ u_out,        // [B, 256]
    float* __restrict__ spk_out)      // [B, 256]
{
    // Transposed, pre-scaled weight tile in LDS: ldsB[n*PITCH + k] = 0.01*x2h[k][n]
    __shared__ float ldsB[N_HID * LDS_PITCH];

    const int tid = threadIdx.x;

    // Stage weights: thread t handles column n = t, all 32 k's (coalesced reads).
    #pragma unroll
    for (int k = 0; k < N_INP; ++k) {
        ldsB[tid * LDS_PITCH + k] = x2h[k * N_HID + tid] * SCALE;
    }
    __syncthreads();

    const int wave = tid >> 5;
    const int lane = tid & 31;
    const int half = lane >> 4;        // 0: lanes 0-15, 1: lanes 16-31
    const int l16  = lane & 15;
    const int row0 = blockIdx.x * ROWS_PER_BLOCK + wave * 16;

    // A-matrix 16x4 f32 fragment layout (ISA 7.12.2):
    //   v0: lanes 0-15 -> K=0, lanes 16-31 -> K=2 ; v1: K=1 / K=3.
    // Lane L holds row m = L&15, columns kb = 4*kk + 2*half, kb+1.
    v2f afrag[8];
    const float* arow = x + (size_t)(row0 + l16) * N_INP;
    #pragma unroll
    for (int kk = 0; kk < 8; ++kk) {
        const int kb = kk * 4 + half * 2;
        afrag[kk] = *(const v2f*)(arow + kb);   // b64 load, 8B aligned
    }

    // 16 output tiles of 16 columns each.
    #pragma unroll 1
    for (int nt = 0; nt < 16; ++nt) {
        const int n = nt * 16 + l16;           // this lane's B/D column
        v8f acc = {};
        #pragma unroll
        for (int kk = 0; kk < 8; ++kk) {
            const int kb = kk * 4 + half * 2;  // mirrored B layout: rows kb, kb+1
            v2f bfrag = *(const v2f*)(&ldsB[n * LDS_PITCH + kb]);
            // 8 args: (neg_a, A, neg_b, B, c_mod, C, reuse_a, reuse_b)
            acc = __builtin_amdgcn_wmma_f32_16x16x4_f32(
                false, afrag[kk], false, bfrag, (short)0, acc, false, false);
        }
        // C/D layout: VGPR j, lanes 0-15 -> M=j, lanes 16-31 -> M=j+8; N=lane&15.
        float* outp = u_out + (size_t)(row0 + 8 * half) * N_HID + nt * 16 + l16;
        #pragma unroll
        for (int j = 0; j < 8; ++j) {
            outp[(size_t)j * N_HID] = acc[j];
        }
    }

    // spk == 0 for the whole batch: zero-fill this block's 128x256 strip (b128 stores).
    float4 z = make_float4(0.f, 0.f, 0.f, 0.f);
    float4* spk4 = (float4*)(spk_out + (size_t)blockIdx.x * ROWS_PER_BLOCK * N_HID);
    #pragma unroll
    for (int i = 0; i < 32; ++i) {             // 8192 float4 per block / 256 threads
        spk4[i * 256 + tid] = z;
    }
}

extern "C" void kernel_launch(void* const* d_in, const int* in_sizes, int n_in,
                              void* d_out, int out_size, void* d_ws, size_t ws_size,
                              hipStream_t stream) {
    const float* x   = (const float*)d_in[0];   // [524288, 32] f32
    const float* x2h = (const float*)d_in[1];   // [32, 256] f32
    // d_in[2] (h2h) is dead: multiplied by an all-zero state in the reference.

    float* u_out   = (float*)d_out;                         // first output
    float* spk_out = u_out + (size_t)B_ROWS * N_HID;        // second output

    dim3 grid(B_ROWS / ROWS_PER_BLOCK);   // 4096 blocks
    dim3 block(256);                      // 8 waves (wave32)
    lsm_gemm_kernel<<<grid, block, 0, stream>>>(x, x2h, u_out, spk_out);
}